// Net_52080773431332
// MI455X (gfx1250) — compile-verified
//
#include <hip/hip_runtime.h>
#include <cstdint>

// ---------------- problem constants (reference: 4096x4096 f32) ----------------
#define EPS_F 1e-7f
#define SIG_OFFSET_F 0.1f
#define SIG_SCALE_F 48.0f

constexpr int W   = 4096;
constexpr int H   = 4096;
constexpr int TW  = 64;              // tile interior width
constexpr int TH  = 32;              // tile interior height
constexpr int PW  = TW + 2;          // 66  (with 1-px halo)
constexpr int PH  = TH + 2;          // 34
constexpr int NT  = 256;             // threads per block = 8 waves (wave32)
constexpr int NBX = W / TW;          // 64
constexpr int NBY = H / TH;          // 128
constexpr int NB  = NBX * NBY;       // 8192 partial sums

typedef unsigned int u32x4 __attribute__((ext_vector_type(4)));
typedef int          i32x8 __attribute__((ext_vector_type(8)));
typedef int          i32x4 __attribute__((ext_vector_type(4)));
typedef float        v8f   __attribute__((ext_vector_type(8)));
typedef float        v2f   __attribute__((ext_vector_type(2)));

#if __has_builtin(__builtin_amdgcn_tensor_load_to_lds) && __has_builtin(__builtin_amdgcn_s_wait_tensorcnt)
#define HAVE_TDM 1
#else
#define HAVE_TDM 0
#endif

#if __has_builtin(__builtin_amdgcn_wmma_f32_16x16x4_f32)
#define HAVE_WMMA_F32X4 1
#else
#define HAVE_WMMA_F32X4 0
#endif

// Low 32 bits of a generic pointer into LDS = byte offset within the
// workgroup's LDS allocation (flat-aperture mapping, ISA 10.2).
__device__ __forceinline__ uint32_t lds_byte_off(const void* p) {
    return (uint32_t)(uintptr_t)p;
}

#if HAVE_TDM
// Issue one 2D TDM tile load: tile_d1 rows of tile_d0 f32 elements,
// row stride stride0 elements, into contiguous LDS at lds_addr.
// Descriptor bitfields per CDNA5 ISA §8.3 (group 0) / §8.4 (group 1).
// This toolchain's builtin takes 6 args:
//   (u32x4 group0, i32x8 group1, i32x4 group2, i32x4 group3, i32x8 extra, i32 cpol)
__device__ __forceinline__ void tdm_load_2d_f32(uint32_t lds_addr, const void* gptr,
                                                uint32_t tile_d0, uint32_t tile_d1,
                                                uint64_t stride0,
                                                uint32_t tensor_d0, uint32_t tensor_d1) {
    uint64_t ga = (uint64_t)(uintptr_t)gptr;
    u32x4 g0;
    g0[0] = 1u;                                                 // count=1 (valid), user mode
    g0[1] = lds_addr;                                           // lds_addr [63:32]
    g0[2] = (uint32_t)(ga & 0xFFFFFFFFull);                     // global_addr [95:64]
    g0[3] = (uint32_t)((ga >> 32) & 0x01FFFFFFull) | (2u << 30); // addr[56:32] | type=2
    i32x8 g1;
    g1[0] = (int)(2u << 16);                                    // data_size=2 -> 4 bytes
    g1[1] = (int)((tensor_d0 & 0xFFFFu) << 16);                 // tensor_dim0[15:0] @ [63:48]
    g1[2] = (int)(((tensor_d0 >> 16) & 0xFFFFu) |
                  ((tensor_d1 & 0xFFFFu) << 16));               // dim0 hi | dim1 lo
    g1[3] = (int)(((tensor_d1 >> 16) & 0xFFFFu) |
                  ((tile_d0 & 0xFFFFu) << 16));                 // dim1 hi | tile_dim0
    g1[4] = (int)(tile_d1 & 0xFFFFu);                           // tile_dim1 | tile_dim2=0
    g1[5] = (int)(uint32_t)(stride0 & 0xFFFFFFFFull);           // tensor_dim0_stride lo
    g1[6] = (int)(uint32_t)((stride0 >> 32) & 0xFFFFull);       // stride hi | dim1_stride lo=0
    g1[7] = 0;
    i32x4 z4 = {0, 0, 0, 0};                                    // groups 2/3 unused (2D tile)
    i32x8 z8 = {0, 0, 0, 0, 0, 0, 0, 0};
    __builtin_amdgcn_tensor_load_to_lds(g0, g1, z4, z4, z8, 0);
}
#endif

__global__ __launch_bounds__(NT) void
tile_stencil_kernel(const float* __restrict__ tmap,
                    const float* __restrict__ limg,
                    float* __restrict__ partial) {
    __shared__ float s_t[PH * PW];         // raw tmap tile, then log in-place
    __shared__ float s_c[PH * PW * 3];     // raw l_img tile (interleaved RGB)
    __shared__ float s_g[PH * PW];         // channel mean
    __shared__ float s_red[NT / 32];

    const int bx  = blockIdx.x;
    const int by  = blockIdx.y;
    const int gx0 = bx * TW - 1;           // top-left of padded tile (image coords)
    const int gy0 = by * TH - 1;
    const int tid = (int)threadIdx.x;

    const bool interior = (gx0 >= 0) && (gy0 >= 0) && (gx0 + PW <= W) && (gy0 + PH <= H);

#if HAVE_TDM
    if (interior) {
        if (tid < 32) {                    // wave 0 only: TDM is per-wave, EXEC-ignored
            tdm_load_2d_f32(lds_byte_off(s_t),
                            tmap + (size_t)gy0 * W + gx0,
                            (uint32_t)PW, (uint32_t)PH,
                            (uint64_t)W, (uint32_t)W, (uint32_t)H);
            tdm_load_2d_f32(lds_byte_off(s_c),
                            limg + ((size_t)gy0 * W + gx0) * 3,
                            (uint32_t)(PW * 3), (uint32_t)PH,
                            (uint64_t)W * 3, (uint32_t)(W * 3), (uint32_t)H);
            __builtin_amdgcn_s_wait_tensorcnt(0);
        }
    } else
#endif
    {
        // Fallback / border path: bounds-checked direct loads, zero halo.
        for (int i = tid; i < PH * PW; i += NT) {
            int r = i / PW, c = i - r * PW;
            int gy = gy0 + r, gx = gx0 + c;
            bool in = (gx >= 0) && (gx < W) && (gy >= 0) && (gy < H);
            s_t[i] = in ? tmap[(size_t)gy * W + gx] : 0.0f;
        }
        for (int i = tid; i < PH * PW * 3; i += NT) {
            int r  = i / (PW * 3);
            int k  = i - r * (PW * 3);
            int c  = k / 3;
            int ch = k - c * 3;
            int gy = gy0 + r, gx = gx0 + c;
            bool in = (gx >= 0) && (gx < W) && (gy >= 0) && (gy < H);
            s_c[i] = in ? limg[((size_t)gy * W + gx) * 3 + ch] : 0.0f;
        }
    }
    __syncthreads();

    // Transform once per element: log(clip(t)) and channel mean g.
    // Reference pads the *transformed* arrays with zeros, so OOB -> 0.
    for (int i = tid; i < PH * PW; i += NT) {
        int r = i / PW, c = i - r * PW;
        int gy = gy0 + r, gx = gx0 + c;
        bool in = (gx >= 0) && (gx < W) && (gy >= 0) && (gy < H);
        float t = s_t[i];
        t = fminf(fmaxf(t, EPS_F), 1.0f);
        s_t[i] = in ? logf(t) : 0.0f;
        float g = (s_c[3 * i + 0] + s_c[3 * i + 1] + s_c[3 * i + 2]) * (1.0f / 3.0f);
        s_g[i] = in ? g : 0.0f;
    }
    __syncthreads();

    // Stencil + fused loss accumulate over the TH x TW interior.
    float acc = 0.0f;
    for (int p = tid; p < TH * TW; p += NT) {
        int r = p / TW + 1;
        int c = p - (r - 1) * TW + 1;
        int i = r * PW + c;
        float dxl = s_t[i + 1]  - s_t[i - 1];
        float dyl = s_t[i - PW] - s_t[i + PW];
        float l2  = dxl * dxl + dyl * dyl;
        float dxg = s_g[i + 1]  - s_g[i - 1];
        float dyg = s_g[i - PW] - s_g[i + PW];
        float mag = sqrtf(dxg * dxg + dyg * dyg);
        float xar = (SIG_OFFSET_F - mag) * SIG_SCALE_F;
        float sig = 1.0f / (1.0f + expf(-xar));
        acc += l2 * sig;
    }

    // Deterministic block reduction: wave shuffle -> LDS -> thread 0.
    for (int off = 16; off > 0; off >>= 1)
        acc += __shfl_down(acc, off, 32);
    if ((tid & 31) == 0) s_red[tid >> 5] = acc;
    __syncthreads();
    if (tid == 0) {
        float s = 0.0f;
        for (int w = 0; w < NT / 32; ++w) s += s_red[w];
        partial[(size_t)by * gridDim.x + bx] = s;
    }
}

// Final reduction of NB partials by a single wave; uses the matrix pipe
// (V_WMMA_F32_16X16X4_F32 with all-ones B) as a 64-way accumulation tree.
__global__ __launch_bounds__(32) void
final_reduce_kernel(const float* __restrict__ partial, float* __restrict__ out, int n) {
    const int lane = (int)threadIdx.x;
#if HAVE_WMMA_F32X4
    v2f b; b[0] = 1.0f; b[1] = 1.0f;       // B = ones(4x16): D[m][j] = row-sum(A[m]) + C
    v8f c = {};                             // running 16x16 accumulator
    for (int i = 0; i < n; i += 64) {
        v2f a;
        a[0] = partial[i + lane];
        a[1] = partial[i + 32 + lane];
        c = __builtin_amdgcn_wmma_f32_16x16x4_f32(false, a, false, b,
                                                  (short)0, c, false, false);
    }
    // One column of D holds the 16 row-sums: lanes L and L+16 carry M=0..7 / 8..15.
    float s = c[0] + c[1] + c[2] + c[3] + c[4] + c[5] + c[6] + c[7];
    s += __shfl_xor(s, 16, 32);             // column total == grand total
#else
    float s = 0.0f;
    for (int i = lane; i < n; i += 32) s += partial[i];
    for (int off = 16; off > 0; off >>= 1) s += __shfl_down(s, off, 32);
#endif
    if (lane == 0) out[0] = s;
}

extern "C" void kernel_launch(void* const* d_in, const int* in_sizes, int n_in,
                              void* d_out, int out_size, void* d_ws, size_t ws_size,
                              hipStream_t stream) {
    const float* tmap = (const float*)d_in[0];  // [H, W]
    const float* limg = (const float*)d_in[1];  // [H, W, 3]
    float* out     = (float*)d_out;             // scalar f32
    float* partial = (float*)d_ws;              // NB = 8192 floats of scratch

    dim3 grid(NBX, NBY);
    tile_stencil_kernel<<<grid, NT, 0, stream>>>(tmap, limg, partial);
    final_reduce_kernel<<<1, 32, 0, stream>>>(partial, out, NB);
}